// FP8ToFP16Converter_43860206027339
// MI455X (gfx1250) — compile-verified
//
#include <hip/hip_runtime.h>
#include <stdint.h>

// Clang extended vector type: required so __builtin_nontemporal_load/store
// accept it (HIP's uint4 is a struct and is rejected by those builtins).
typedef uint32_t u32x4 __attribute__((ext_vector_type(4)));

#define F32_ONE_BITS 0x3F800000u

// FP8(E4M3 bit-vector, 8 floats of 0.0/1.0) -> FP16 bit-vector (16 floats).
// Pure streaming: 32 B in, 64 B out per element; NT hints + prefetch.
__global__ __launch_bounds__(256) void fp8_to_fp16_bits_kernel(
    const uint32_t* __restrict__ in,   // float bit patterns: 0x0 or 0x3F800000
    uint32_t*       __restrict__ out,  // write float bit patterns
    int n_elems)
{
    const int i = blockIdx.x * blockDim.x + threadIdx.x;
    if (i >= n_elems) return;

    const size_t ibase = (size_t)i * 8u;

    // gfx1250 prefetch path: pull the read stream ~2 MiB ahead into cache.
    // Guarded so the prefetch address stays inside the input allocation.
    if (i + 65536 < n_elems)
        __builtin_prefetch(in + ibase + (size_t)65536 * 8u, 0, 0);

    const u32x4* inv = (const u32x4*)(in + ibase);
    // Non-temporal 128-bit loads: zero-reuse stream, don't rinse L2.
    u32x4 a = __builtin_nontemporal_load(inv);      // s, e3, e2, e1
    u32x4 b = __builtin_nontemporal_load(inv + 1);  // e0, m2, m1, m0

    // 1.0f == 0x3F800000 -> bit 23 is the value; 0.0f -> 0.
    const uint32_t s  = (a.x >> 23) & 1u;
    const uint32_t e3 = (a.y >> 23) & 1u;
    const uint32_t e2 = (a.z >> 23) & 1u;
    const uint32_t e1 = (a.w >> 23) & 1u;
    const uint32_t e0 = (b.x >> 23) & 1u;
    const uint32_t m2 = (b.y >> 23) & 1u;
    const uint32_t m1 = (b.z >> 23) & 1u;
    const uint32_t m0 = (b.w >> 23) & 1u;

    const uint32_t e = (e3 << 3) | (e2 << 2) | (e1 << 1) | e0;  // 0..15
    const uint32_t m = (m2 << 2) | (m1 << 1) | m0;              // 0..7

    // Normal path (e != 0; FP8 NaN rides through the 5-bit adder like the ref):
    //   exp = e + 8 (rebias 7 -> 15), mant = m2m1m0 in mant[9:7].
    const uint32_t expN  = e + 8u;       // <= 23, fits 5 bits
    const uint32_t mantN = m << 7;

    // Subnormal path (e == 0, m != 0): normalize by leading-one position p.
    //   p=2 -> exp 8, mant {m1,m0}<<8 ; p=1 -> exp 7, mant m0<<9 ; p=0 -> exp 6, mant 0.
    const uint32_t p     = (m >= 4u) ? 2u : ((m >= 2u) ? 1u : 0u);
    const uint32_t expS  = 6u + p;
    const uint32_t mantS = (m << (10u - p)) & 0x3FFu;

    const bool normal = (e != 0u);
    const bool zero   = (e == 0u) & (m == 0u);

    const uint32_t exp5   = zero ? 0u : (normal ? expN  : expS);
    const uint32_t mant10 = zero ? 0u : (normal ? mantN : mantS);

    // FP16 bit image, MSB-first fan-out matches the reference's channel order:
    // out[0]=s, out[1..5]=exp[4:0], out[6..15]=mant[9:0].
    const uint32_t bits = (s << 15) | (exp5 << 10) | mant10;

    uint32_t w[16];
#pragma unroll
    for (int j = 0; j < 16; ++j)
        w[j] = ((bits >> (15 - j)) & 1u) ? F32_ONE_BITS : 0u;

    u32x4* outv = (u32x4*)(out + (size_t)i * 16u);
#pragma unroll
    for (int v = 0; v < 4; ++v) {
        u32x4 o;
        o.x = w[4 * v + 0];
        o.y = w[4 * v + 1];
        o.z = w[4 * v + 2];
        o.w = w[4 * v + 3];
        // Non-temporal 128-bit stores: 1 GiB write stream, no reuse.
        __builtin_nontemporal_store(o, outv + v);
    }
}

extern "C" void kernel_launch(void* const* d_in, const int* in_sizes, int n_in,
                              void* d_out, int out_size, void* d_ws, size_t ws_size,
                              hipStream_t stream) {
    (void)n_in; (void)out_size; (void)d_ws; (void)ws_size;

    const uint32_t* in  = (const uint32_t*)d_in[0];  // float32 bit patterns
    uint32_t*       out = (uint32_t*)d_out;          // float32 bit patterns

    const int n_elems = in_sizes[0] / 8;             // 8192*2048 = 16,777,216
    const int block   = 256;                         // 8 wave32s
    const int grid    = (n_elems + block - 1) / block;

    fp8_to_fp16_bits_kernel<<<grid, block, 0, stream>>>(in, out, n_elems);
}